// SE2CGNonLinearity_10282151706947
// MI455X (gfx1250) — compile-verified
//
#include <hip/hip_runtime.h>
#include <hip/hip_bf16.h>

// SE(2) circle-group CG nonlinearity:
// out[l] = sum_m w[l,m,o] * x[b,m,o,p] * x[b,(l-m)%8,o,p]   (complex)
//
// Symmetry-reduced WMMA mapping: u[m,n]=u[n,m], so contract over the 36
// unordered pairs q={m,n} with paired weights C[l,q] = w[l,m]+w[l,n]
// (w[l,m] on the diagonal), gated by the CG rule m+n==l (mod 8).
// D(16x16) = A(16x72) x B(72x16) per 16-pixel tile, 18 x V_WMMA_F32_16X16X4_F32:
//   A rows 0-7  = [ Cr | -Ci ],  rows 8-15 = [ Ci | Cr ]
//   B rows 0-35 = u_r[q],        rows 36-71 = u_i[q]
//   D rows 0-7  = out_real(l),   rows 8-15  = out_imag(l)

typedef __attribute__((ext_vector_type(2))) float v2f;
typedef __attribute__((ext_vector_type(8))) float v8f;

#if defined(__HIP_DEVICE_COMPILE__)
#if !__has_builtin(__builtin_amdgcn_wmma_f32_16x16x4_f32)
#error "wmma_f32_16x16x4_f32 builtin not available on this toolchain (device pass)"
#endif
#endif

#define LB 8      // freq cutoff L
#define OB 64     // channels
#define BB 8      // batch
#define HWB 4096  // H*W
#define NQ 36     // unordered pairs {m,n}, m<=n
#define KTOT 72   // 36 u_r + 36 u_i
#define NSTEP 18  // KTOT / 4
#define ASTRIDE 76  // 72 K + 4 pad -> bank = (12*row + k) % 64, conflict-free

// Unordered pair tables: q -> (m, n), row-major upper triangle incl. diagonal.
constexpr int QM[NQ] = {0,0,0,0,0,0,0,0, 1,1,1,1,1,1,1, 2,2,2,2,2,2,
                        3,3,3,3,3, 4,4,4,4, 5,5,5, 6,6, 7};
constexpr int QN[NQ] = {0,1,2,3,4,5,6,7, 1,2,3,4,5,6,7, 2,3,4,5,6,7,
                        3,4,5,6,7, 4,5,6,7, 5,6,7, 6,7, 7};

__global__ __launch_bounds__(128) void se2cg_wmma_kernel(
    const float* __restrict__ x_real, const float* __restrict__ x_imag,
    const float* __restrict__ w_real, const float* __restrict__ w_imag,
    float* __restrict__ out)
{
    __shared__ float Ash[16 * ASTRIDE];

    const int chunk = blockIdx.x;   // 8 chunks of 512 pixels
    const int o     = blockIdx.y;   // 64
    const int b     = blockIdx.z;   // 8
    const int tid   = threadIdx.x;
    const int lane  = tid & 31;
    const int wave  = tid >> 5;     // 4 waves
    const int half  = lane >> 4;    // half-wave id (K-row pair / real-imag out)
    const int p     = lane & 15;    // A-row M, B-column N (pixel), D-column N

    // ---- Build A(16 x 72) in LDS once per block (depends only on o) ----
    // 1152 entries, 9 per thread. Pair index inverted with a tiny loop.
    #pragma unroll
    for (int j = 0; j < 9; ++j) {
        int idx = tid + 128 * j;          // 0..1151
        int row = idx / KTOT;             // 0..15
        int k   = idx % KTOT;             // 0..71
        int ri  = (k >= NQ) ? 1 : 0;      // 0: u_r half, 1: u_i half
        int q   = k - NQ * ri;            // 0..35
        int m = 0, rem = q;
        while (rem >= 8 - m) { rem -= 8 - m; ++m; }
        int n = m + rem;                  // m <= n
        int l = row & 7;
        float cr = w_real[(l * 8 + m) * OB + o];
        float ci = w_imag[(l * 8 + m) * OB + o];
        if (m != n) {
            cr += w_real[(l * 8 + n) * OB + o];
            ci += w_imag[(l * 8 + n) * OB + o];
        }
        // rows 0-7: [Cr | -Ci]; rows 8-15: [Ci | Cr]
        float val = (row < 8) ? (ri ? -ci : cr) : (ri ? cr : ci);
        bool match = (((m + n) & 7) == l);      // CG selection rule
        Ash[row * ASTRIDE + k] = match ? val : 0.0f;
    }
    __syncthreads();

    const long mstride = (long)OB * HWB;  // stride between freq indices m in x
    const long xbase   = ((long)b * LB * OB + o) * HWB;

    // ---- 8 tiles of 16 pixels per wave ----
    #pragma unroll 1
    for (int tile = 0; tile < 8; ++tile) {
        const int pix = chunk * 512 + wave * 128 + tile * 16 + p;

        // Each lane loads the 8 complex x values of its pixel
        float xr[8], xi[8];
        #pragma unroll
        for (int m = 0; m < 8; ++m) {
            long off = xbase + (long)m * mstride + pix;
            xr[m] = x_real[off];
            xi[m] = x_imag[off];
        }

        v8f acc = {0.f, 0.f, 0.f, 0.f, 0.f, 0.f, 0.f, 0.f};

        // 18 chained V_WMMA_F32_16X16X4_F32, K = 4t..4t+3.
        // Steps 0-8 consume u_r pairs, steps 9-17 u_i pairs (36 = 9*4).
        #pragma unroll
        for (int t = 0; t < NSTEP; ++t) {
            // A fragment: lane(half,p) holds A[p][4t+2*half], A[p][4t+2*half+1]
            const int k0 = 4 * t + 2 * half;
            v2f a = *(const v2f*)(Ash + p * ASTRIDE + k0);

            // B fragment: u for this lane's two K rows. All (m,n) compile-time;
            // select by half to avoid dynamic register indexing.
            const int ri = (t >= 9) ? 1 : 0;
            const int s  = t - 9 * ri;        // base pair index group
            float u[4];
            #pragma unroll
            for (int kk = 0; kk < 4; ++kk) {
                const int q = 4 * s + kk;     // compile-time
                const int m = QM[q];
                const int n = QN[q];
                u[kk] = ri ? __builtin_fmaf(xr[m], xi[n],  xi[m] * xr[n])   // u_i
                           : __builtin_fmaf(xr[m], xr[n], -(xi[m] * xi[n])); // u_r
            }
            v2f bb;
            bb.x = half ? u[2] : u[0];
            bb.y = half ? u[3] : u[1];

            acc = __builtin_amdgcn_wmma_f32_16x16x4_f32(
                false, a, false, bb, (short)0, acc, false, false);
        }

        // D: VGPR j, lanes 0-15 -> row j (real, l=j); lanes 16-31 -> row j+8 (imag, l=j)
        #pragma unroll
        for (int j = 0; j < 8; ++j) {
            long oidx = ((((long)half * BB + b) * LB + j) * OB + o) * HWB + pix;
            out[oidx] = acc[j];
        }
    }
}

extern "C" void kernel_launch(void* const* d_in, const int* in_sizes, int n_in,
                              void* d_out, int out_size, void* d_ws, size_t ws_size,
                              hipStream_t stream) {
    const float* x_real = (const float*)d_in[0];
    const float* x_imag = (const float*)d_in[1];
    const float* w_real = (const float*)d_in[2];
    const float* w_imag = (const float*)d_in[3];
    // d_in[4] = CG mask: selection rule is hard-coded in the kernel.
    float* out = (float*)d_out;

    dim3 grid(8, OB, BB);   // pixel-chunk, o, b
    dim3 block(128);        // 4 wave32
    se2cg_wmma_kernel<<<grid, block, 0, stream>>>(x_real, x_imag, w_real, w_imag, out);
}